// ModelNew_25056839205311
// MI455X (gfx1250) — compile-verified
//
#include <hip/hip_runtime.h>
#include <hip/hip_bf16.h>

// ---------------------------------------------------------------------------
// MoE token permutation for MI455X (gfx1250, wave32).
// Shapes specialized to the reference: M=8192, K=4096, TOPK=8, G=64, BM=128.
// n_slots = 65536, capacity = 73728. Row = 8KB bf16.
// Main cost: ~604MB of stores -> HBM bound (~26us @ 23.3 TB/s). Source rows
// (64MiB) fit in the 192MB L2, so gather reads are mostly L2 hits; output
// stores use non-temporal policy to avoid evicting the source from L2.
// Big copy uses CDNA5 async global<->LDS b128 instructions (ASYNCcnt path).
// ---------------------------------------------------------------------------

#define AS1 __attribute__((address_space(1)))
#define AS3 __attribute__((address_space(3)))

typedef __attribute__((__vector_size__(4 * sizeof(int)))) int v4i;

#if defined(__HIP_DEVICE_COMPILE__) && \
    __has_builtin(__builtin_amdgcn_global_load_async_to_lds_b128) && \
    __has_builtin(__builtin_amdgcn_global_store_async_from_lds_b128)
#define USE_ASYNC_LDS 1
#else
#define USE_ASYNC_LDS 0
#endif

// gfx12+ CPol: TH[2:0], SCOPE[4:3]. TH_NT == 1 (non-temporal).
#define CPOL_RT 0
#define CPOL_NT 1

__device__ __forceinline__ void moe_wait_async_zero() {
#if defined(__HIP_DEVICE_COMPILE__)
#if __has_builtin(__builtin_amdgcn_s_wait_asynccnt)
    __builtin_amdgcn_s_wait_asynccnt(0);
#else
    asm volatile("s_wait_asynccnt 0" ::: "memory");
#endif
#endif
}

__device__ __forceinline__ unsigned moe_ballot32(bool p) {
#if defined(__HIP_DEVICE_COMPILE__) && __has_builtin(__builtin_amdgcn_ballot_w32)
    return (unsigned)__builtin_amdgcn_ballot_w32(p);
#else
    return (unsigned)__ballot(p);
#endif
}

static constexpr int G64     = 64;   // num_groups
static constexpr int TOPK8   = 8;    // topk
static constexpr int BM128   = 128;  // block_m
static constexpr int HBLK    = 1024; // slots per histogram/scatter block

// --- big row copy: one 256-thread block per 8KB row ------------------------
// (defined first so the disasm snippet shows the async-LDS instructions)
__global__ __launch_bounds__(256) void moe_copy_rows(
    const __hip_bfloat16* __restrict__ hidden,
    const int* __restrict__ row_token,
    __hip_bfloat16* __restrict__ out, int K) {
    __shared__ __align__(16) unsigned char smem[8192];
    const int r = blockIdx.x;
    const int t = threadIdx.x;
    const unsigned row_bytes = (unsigned)K * 2u;  // 8192
    char* dst = (char*)out + (size_t)r * row_bytes;
    const int tok = row_token[r];  // uniform per block
    if (tok < 0) {
        const v4i z = {0, 0, 0, 0};
        for (unsigned off = (unsigned)t * 16u; off < row_bytes; off += 4096u)
            __builtin_nontemporal_store(z, reinterpret_cast<v4i*>(dst + off));
        return;
    }
    const char* src = (const char*)hidden + (size_t)tok * row_bytes;
#if USE_ASYNC_LDS
    // CDNA5 async DMA-style path: global -> LDS -> global, b128 per lane.
    // Loads: RT (keep 64MiB hidden_states resident in the 192MB L2).
    for (unsigned off = (unsigned)t * 16u; off < row_bytes; off += 4096u)
        __builtin_amdgcn_global_load_async_to_lds_b128(
            (AS1 v4i*)(src + off), (AS3 v4i*)(smem + off), 0, CPOL_RT);
    moe_wait_async_zero();  // each lane stores only bytes it loaded: no barrier needed
    // Stores: NT (604MB output stream must not evict the source from L2).
    for (unsigned off = (unsigned)t * 16u; off < row_bytes; off += 4096u)
        __builtin_amdgcn_global_store_async_from_lds_b128(
            (AS1 v4i*)(dst + off), (AS3 v4i*)(smem + off), 0, CPOL_NT);
    // outstanding async stores are drained by S_ENDPGM's implicit wait-idle
#else
    for (unsigned off = (unsigned)t * 16u; off < row_bytes; off += 4096u) {
        const v4i v = *reinterpret_cast<const v4i*>(src + off);
        __builtin_nontemporal_store(v, reinterpret_cast<v4i*>(dst + off));
    }
#endif
}

// --- 1) per-block expert histogram ----------------------------------------
__global__ __launch_bounds__(HBLK) void moe_h1_hist(
    const int* __restrict__ flat, int* __restrict__ blockHist) {
    __shared__ int h[G64];
    const int t = threadIdx.x, b = blockIdx.x;
    if (t < G64) h[t] = 0;
    __syncthreads();
    atomicAdd(&h[flat[(size_t)b * HBLK + t] & (G64 - 1)], 1);
    __syncthreads();
    if (t < G64) blockHist[b * G64 + t] = h[t];
}

// --- 2) cross-block scan + starts/offsets ---------------------------------
__global__ __launch_bounds__(64) void moe_h2_scan(
    const int* __restrict__ blockHist, int nBlocks,
    int* __restrict__ blockBase, int* __restrict__ counts,
    int* __restrict__ starts, int* __restrict__ offsets32,
    long long* __restrict__ offsets_out) {
    const int g = threadIdx.x;  // 64 threads, one per expert
    int acc = 0;
    for (int b = 0; b < nBlocks; ++b) {
        const int c = blockHist[b * G64 + g];
        blockBase[b * G64 + g] = acc;
        acc += c;
    }
    counts[g] = acc;
    __syncthreads();
    if (g == 0) {
        int s = 0, o = 0;
        for (int i = 0; i < G64; ++i) {
            starts[i] = s; offsets32[i] = o; offsets_out[i] = (long long)o;
            s += counts[i];
            o += ((counts[i] + BM128 - 1) / BM128) * BM128;
        }
        starts[G64] = s; offsets32[G64] = o; offsets_out[G64] = (long long)o;
    }
}

// --- 3) stable scatter: order[pos] = slot ----------------------------------
__global__ __launch_bounds__(HBLK) void moe_h3_scatter(
    const int* __restrict__ flat, const int* __restrict__ blockBase,
    const int* __restrict__ starts, int* __restrict__ order) {
    __shared__ int waveOff[HBLK / 32][G64];  // 32 waves x 64 experts = 8KB
    const int b = blockIdx.x, t = threadIdx.x;
    const int w = t >> 5, lane = t & 31;
    const int slot = b * HBLK + t;
    const int e = flat[slot] & (G64 - 1);
    const unsigned ltmask = (1u << lane) - 1u;
    int myrank = 0;
    for (int g = 0; g < G64; ++g) {
        const unsigned m = moe_ballot32(e == g);  // stable: lane order == slot order
        if (e == g) myrank = __popc(m & ltmask);
        if (lane == 0) waveOff[w][g] = __popc(m);
    }
    __syncthreads();
    if (t < G64) {  // exclusive prefix across waves, per expert
        int acc = 0;
        for (int w2 = 0; w2 < HBLK / 32; ++w2) {
            const int c = waveOff[w2][t];
            waveOff[w2][t] = acc;
            acc += c;
        }
    }
    __syncthreads();
    const int pos = starts[e] + blockBase[b * G64 + e] + waveOff[w][e] + myrank;
    order[pos] = slot;
}

// --- 4) per-row metadata: sorted_ids, m_indices, row_token -----------------
__global__ __launch_bounds__(256) void moe_meta(
    const int* __restrict__ offsets32, const int* __restrict__ starts,
    const int* __restrict__ counts, const int* __restrict__ order,
    long long* __restrict__ sorted_ids, int* __restrict__ m_indices,
    int* __restrict__ row_token, int cap, int n_slots) {
    const int r = blockIdx.x * blockDim.x + threadIdx.x;
    if (r >= cap) return;
    // upper_bound over offsets[0..64] (== searchsorted side='right'), then -1
    int l = 0, h = G64 + 1;
    while (l < h) {
        const int m = (l + h) >> 1;
        if (offsets32[m] <= r) l = m + 1; else h = m;
    }
    int g = l - 1;
    g = g < 0 ? 0 : (g > G64 - 1 ? G64 - 1 : g);
    const bool valid = r < offsets32[G64];
    const int c = counts[g];
    const int i_in = r - offsets32[g];
    int within = i_in < c - 1 ? i_in : c - 1;
    int idx = starts[g] + within;
    idx = idx < 0 ? 0 : (idx >= n_slots ? n_slots - 1 : idx);
    const int slot = (c > 0) ? order[idx] : 0;
    const int token = slot / TOPK8;
    sorted_ids[r] = valid ? (long long)slot : 0ll;
    m_indices[r]  = valid ? g : -1;
    row_token[r]  = valid ? token : -1;
}

// ---------------------------------------------------------------------------
extern "C" void kernel_launch(void* const* d_in, const int* in_sizes, int n_in,
                              void* d_out, int out_size, void* d_ws, size_t ws_size,
                              hipStream_t stream) {
    const __hip_bfloat16* hidden = (const __hip_bfloat16*)d_in[0];
    const int* topk_ids          = (const int*)d_in[1];

    const int n_slots = in_sizes[1];              // M*TOPK = 65536
    const int Mrows   = n_slots / TOPK8;          // 8192
    const int K       = in_sizes[0] / Mrows;      // 4096
    const int cap     = n_slots + G64 * BM128;    // 73728
    const int nBlocks = n_slots / HBLK;           // 64

    // workspace layout (int32 units)
    int* ws        = (int*)d_ws;
    int* blockHist = ws;                          // 64*64
    int* blockBase = blockHist + nBlocks * G64;   // 64*64
    int* counts    = blockBase + nBlocks * G64;   // 64
    int* starts    = counts + G64;                // 65
    int* offsets32 = starts + (G64 + 1);          // 65
    int* order     = offsets32 + (G64 + 1);       // n_slots
    int* row_token = order + n_slots;             // cap

    // output layout: out[cap*K] bf16 | sorted_ids[cap] i64 | m_indices[cap] i32 | offsets[G+1] i64
    char* outb = (char*)d_out;
    __hip_bfloat16* out      = (__hip_bfloat16*)outb;
    long long* sorted_ids    = (long long*)(outb + (size_t)cap * (size_t)K * 2u);
    int* m_indices           = (int*)((char*)sorted_ids + (size_t)cap * 8u);
    long long* offsets_out   = (long long*)((char*)m_indices + (size_t)cap * 4u);

    moe_h1_hist<<<nBlocks, HBLK, 0, stream>>>(topk_ids, blockHist);
    moe_h2_scan<<<1, 64, 0, stream>>>(blockHist, nBlocks, blockBase, counts,
                                      starts, offsets32, offsets_out);
    moe_h3_scatter<<<nBlocks, HBLK, 0, stream>>>(topk_ids, blockBase, starts, order);
    moe_meta<<<(cap + 255) / 256, 256, 0, stream>>>(offsets32, starts, counts, order,
                                                    sorted_ids, m_indices, row_token,
                                                    cap, n_slots);
    moe_copy_rows<<<cap, 256, 0, stream>>>(hidden, row_token, out, K);
}